// DigitCaps_12214886990601
// MI455X (gfx1250) — compile-verified
//
#include <hip/hip_runtime.h>

// DigitCaps dynamic routing, fused single kernel for gfx1250 (MI455X).
//
// Decomposition: workgroup = (16-batch block, one output capsule o); grid 32x10.
// u_hat[b,n,o,d] tiles (16b x 16d, K=8) computed by V_WMMA_F32_16X16X4_F32 pairs.
// Pass 1 (b_ij = 0 -> uniform softmax): s_j = mean_n u_hat, carried directly in the
//   WMMA accumulator across the whole n-stream (pure matrix-core loop, no VALU).
// Passes 2-3: flash-style online softmax over n in registers (lane owns fixed (b,d)),
//   base-2 exp (v_exp_f32 native), cross-wave merge via LDS, squash via lane shuffles.
// All inputs (~25 MB) are L2-resident on MI455X (192 MB L2), so recomputing u_hat
// each pass beats materializing the 377 MB u_hat tensor in HBM.
// n-loops use a compile-time trip count (1152/8 = 144) and base+i*const indexing so
// the compiler unrolls and folds strides into vmem instruction offsets.

typedef float v2f __attribute__((ext_vector_type(2)));
typedef float v8f __attribute__((ext_vector_type(8)));

#define BB 512
#define NN 1152
#define OO 10
#define II 8
#define DD 16
#define NWAVES 8                    // 256 threads = 8 wave32
#define NTILES (NN / NWAVES)        // 144 n-tiles per wave, exact
#define WSTRIDE (OO * DD * II)      // W elements per n
#define LOG2E 1.4426950408889634f

// sum across each 16-lane half of the wave (masks < 16 stay within the half)
__device__ __forceinline__ float red16(float v) {
#pragma unroll
  for (int m = 1; m < 16; m <<= 1) v += __shfl_xor(v, m, 32);
  return v;
}

__global__ __launch_bounds__(NWAVES * 32)
void digitcaps_route_kernel(const float* __restrict__ x,   // (B,N,I)
                            const float* __restrict__ W,   // (N,O,D,I)
                            float* __restrict__ out) {     // (B,O,D)
  const int tid  = threadIdx.x;
  const int lane = tid & 31;
  const int wave = tid >> 5;
  const int lo   = lane & 15;   // d index (B/C/D matrix column), b-row for A
  const int hi   = lane >> 4;   // K-group selector for A/B, b-row-half for C/D
  const int b0   = blockIdx.x * 16;
  const int o    = blockIdx.y;

  // per-wave partials: [wave][vgpr r][lane] -> one (b,d) slot each
  __shared__ float sm[NWAVES][8][32];
  __shared__ float sl[NWAVES][8][32];
  __shared__ float ss[NWAVES][8][32];

  // Per-lane bases for this wave's first n-tile (n = wave):
  // A: row M = lo (batch), K = 2*hi + {0,1} (+4 for 2nd WMMA)
  // B: col N = lo (d),     K = 2*hi + {0,1} (+4 for 2nd WMMA)
  const float* xa0 = x + ((size_t)(b0 + lo) * NN + wave) * II + 2 * hi;
  const float* wb0 = W + (((size_t)wave * OO + o) * DD + lo) * II + 2 * hi;
  const float ninf = -__builtin_inff();

  float S[8];     // cumulative routing scale: b_ij = u_hat * S  (per b-row r)
  float vcur[8];  // current v_j for this lane's (b,d) slots

  // ---------------- Pass 1: s_j = mean_n u_hat  (pure WMMA stream) ----------------
  {
    v8f acc = {};  // running sum over this wave's n-slice, kept in the accumulator
#pragma unroll 4
    for (int i = 0; i < NTILES; ++i) {
      const float* xa = xa0 + (size_t)i * (NWAVES * II);
      const float* wb = wb0 + (size_t)i * (NWAVES * WSTRIDE);
      v2f a_lo = *(const v2f*)(xa);
      v2f a_hi = *(const v2f*)(xa + 4);
      v2f b_lo = *(const v2f*)(wb);
      v2f b_hi = *(const v2f*)(wb + 4);
      __builtin_prefetch(xa + 4 * NWAVES * II, 0, 0);
      __builtin_prefetch(wb + 4 * NWAVES * WSTRIDE, 0, 0);
      acc = __builtin_amdgcn_wmma_f32_16x16x4_f32(false, a_lo, false, b_lo,
                                                  (short)0, acc, false, false);
      acc = __builtin_amdgcn_wmma_f32_16x16x4_f32(false, a_hi, false, b_hi,
                                                  (short)0, acc, false, false);
    }
#pragma unroll
    for (int r = 0; r < 8; ++r) ss[wave][r][lane] = acc[r];
    __syncthreads();

#pragma unroll
    for (int r = 0; r < 8; ++r) {
      float sum = 0.0f;
      for (int w = 0; w < NWAVES; ++w) sum += ss[w][r][lane];
      float sj = sum * (1.0f / NN);                    // uniform c_ij = 1/N
      float sn = red16(sj * sj);                       // |s_j|^2 over d
      float v  = sn * sj / ((1.0f + sn) * sqrtf(sn));  // squash (faithful formula)
      vcur[r]  = v;
      S[r]     = red16(v);                             // b_ij += u_hat * sum_d(v_j)
    }
    __syncthreads();
  }

  // ---------------- Passes 2..3: online softmax over n ----------------
  for (int it = 0; it < 2; ++it) {
    float Sl2[8];  // S scaled to base-2 domain: exp(u*S) == exp2(u*S*log2(e))
#pragma unroll
    for (int r = 0; r < 8; ++r) Sl2[r] = S[r] * LOG2E;

    float m[8], l[8], s[8];
#pragma unroll
    for (int r = 0; r < 8; ++r) { m[r] = ninf; l[r] = 0.0f; s[r] = 0.0f; }

#pragma unroll 2
    for (int i = 0; i < NTILES; ++i) {
      const float* xa = xa0 + (size_t)i * (NWAVES * II);
      const float* wb = wb0 + (size_t)i * (NWAVES * WSTRIDE);
      v2f a_lo = *(const v2f*)(xa);
      v2f a_hi = *(const v2f*)(xa + 4);
      v2f b_lo = *(const v2f*)(wb);
      v2f b_hi = *(const v2f*)(wb + 4);
      __builtin_prefetch(xa + 2 * NWAVES * II, 0, 0);
      __builtin_prefetch(wb + 2 * NWAVES * WSTRIDE, 0, 0);

      v8f acc = {};
      acc = __builtin_amdgcn_wmma_f32_16x16x4_f32(false, a_lo, false, b_lo,
                                                  (short)0, acc, false, false);
      acc = __builtin_amdgcn_wmma_f32_16x16x4_f32(false, a_hi, false, b_hi,
                                                  (short)0, acc, false, false);
      // acc[r] = u_hat[b0 + r + 8*hi, n, o, lo]

#pragma unroll
      for (int r = 0; r < 8; ++r) {
        float u  = acc[r];
        float z2 = u * Sl2[r];           // routing logit in log2 domain
        float mn = fmaxf(m[r], z2);
        float sc = exp2f(m[r] - mn);     // rescale old accumulators (v_exp_f32)
        float e  = exp2f(z2 - mn);
        l[r] = l[r] * sc + e;            // sum of exp
        s[r] = s[r] * sc + e * u;        // sum of exp * u_hat
        m[r] = mn;
      }
    }

    // publish per-wave partials and merge across waves
#pragma unroll
    for (int r = 0; r < 8; ++r) {
      sm[wave][r][lane] = m[r];
      sl[wave][r][lane] = l[r];
      ss[wave][r][lane] = s[r];
    }
    __syncthreads();

#pragma unroll
    for (int r = 0; r < 8; ++r) {
      float M = ninf, L = 0.0f, Sv = 0.0f;
      for (int w = 0; w < NWAVES; ++w) {
        float mw = sm[w][r][lane];
        float lw = sl[w][r][lane];
        float sw = ss[w][r][lane];
        float Mn = fmaxf(M, mw);
        float aM = exp2f(M - Mn);
        float aw = exp2f(mw - Mn);
        L  = L * aM + lw * aw;
        Sv = Sv * aM + sw * aw;
        M  = Mn;
      }
      float sj = Sv / L;                               // s_j[b,o,d]
      float sn = red16(sj * sj);                       // |s_j|^2 over d
      float v  = sn * sj / ((1.0f + sn) * sqrtf(sn));  // squash
      vcur[r]  = v;
      S[r]    += red16(v);                             // accumulate routing scale
    }
    __syncthreads();  // LDS reused next pass
  }

  // every wave holds identical v_j; wave 0 writes the (16b x 16d) result tile
  if (wave == 0) {
#pragma unroll
    for (int r = 0; r < 8; ++r) {
      int bg = b0 + r + 8 * hi;
      out[((size_t)bg * OO + o) * DD + lo] = vcur[r];
    }
  }
}

extern "C" void kernel_launch(void* const* d_in, const int* in_sizes, int n_in,
                              void* d_out, int out_size, void* d_ws, size_t ws_size,
                              hipStream_t stream) {
  const float* x = (const float*)d_in[0];   // (512,1152,8) f32
  const float* W = (const float*)d_in[1];   // (1152,10,16,8) f32
  float* out = (float*)d_out;               // (512,10,16) f32
  (void)in_sizes; (void)n_in; (void)out_size; (void)d_ws; (void)ws_size;

  dim3 grid(BB / 16, OO);                   // 32 x 10 workgroups
  digitcaps_route_kernel<<<grid, NWAVES * 32, 0, stream>>>(x, W, out);
}